// HL_41996190220467
// MI455X (gfx1250) — compile-verified
//
#include <hip/hip_runtime.h>

// 4-D multilinear interpolation on a per-point 8^4 lattice.
// Each wave32 handles 32 points; two groups of 16 points each are reduced
// through chained V_WMMA_F32_16X16X4_F32 (K = 16 corners in 4 steps),
// with the per-point dot products landing on the D-matrix diagonal.

typedef __attribute__((ext_vector_type(2))) float v2f;
typedef __attribute__((ext_vector_type(8))) float v8f;

#define VOLUME 4096   // 8*8*8*8

__global__ __launch_bounds__(256) void HL_interp4d_wmma(
    const float* __restrict__ coords,   // [B, 4]
    const float* __restrict__ mesh,     // [B, 4096]
    float* __restrict__ out)            // [B]
{
    const int tid      = blockIdx.x * 256 + (int)threadIdx.x;
    const int lane     = (int)threadIdx.x & 31;
    const int waveBase = tid - lane;    // first point handled by this wave

    // ---- phase 1: per-lane point setup (point == global thread id) ----
    const float4 c4 = *reinterpret_cast<const float4*>(coords + 4 * (size_t)tid);
    const float cc[4] = {c4.x, c4.y, c4.z, c4.w};
    const int strides[4] = {512, 64, 8, 1};
    int   base = 0;
    float fr[4];
#pragma unroll
    for (int d = 0; d < 4; ++d) {
        float c = cc[d] * 7.0f;               // scale by (mesh-1)
        int   i = (int)c;                     // trunc == floor (c >= 0)
        i = i < 0 ? 0 : (i > 6 ? 6 : i);      // clamp to [0, s-2]
        fr[d] = c - (float)i;
        base += i * strides[d];
    }

    // ---- phase 2: two 16-point groups per wave through WMMA ----
#pragma unroll
    for (int g = 0; g < 2; ++g) {
        const int p = (lane & 15) + 16 * g;   // wave-local point this lane serves
        const int n = waveBase + p;           // global point id
        // broadcast point p's fracs + base cell index to its serving lanes
        const float f0 = __shfl(fr[0], p);
        const float f1 = __shfl(fr[1], p);
        const float f2 = __shfl(fr[2], p);
        const float f3 = __shfl(fr[3], p);
        const int   pb = __shfl(base,  p);
        const float* row = mesh + (size_t)n * VOLUME + pb;

        v8f acc = {};                         // C starts at zero
        const int khalf = (lane >> 4) << 1;   // lanes 0-15: K%4 in {0,1}; 16-31: {2,3}
#pragma unroll
        for (int s = 0; s < 4; ++s) {
            const int k = 4 * s + khalf;      // even corner -> VGPR0, k+1 -> VGPR1
            // weights: prod_d (bit ? frac : 1-frac); corners k,k+1 differ in bit0
            const float w0 = (k & 8) ? f0 : (1.0f - f0);
            const float w1 = (k & 4) ? f1 : (1.0f - f1);
            const float w2 = (k & 2) ? f2 : (1.0f - f2);
            const float w012 = w0 * w1 * w2;
            v2f A;
            A.x = w012 * (1.0f - f3);
            A.y = w012 * f3;
            // values: corners k and k+1 are adjacent in memory (stride-1 dim)
            const int off = ((k >> 3) & 1) * 512 + ((k >> 2) & 1) * 64 + ((k >> 1) & 1) * 8;
            v2f B;
            B.x = row[off];
            B.y = row[off + 1];
            // D = A(16x4) x B(4x16) + C ; diagonal accumulates per-point dots
            acc = __builtin_amdgcn_wmma_f32_16x16x4_f32(
                /*neg_a=*/false, A, /*neg_b=*/false, B,
                /*c_mod=*/(short)0, acc, /*reuse_a=*/false, /*reuse_b=*/false);
        }

        // ---- diagonal extraction ----
        // (m,m) lives in: lanes 0-7 -> VGPR=lane ; lanes 24-31 -> VGPR=lane-24
        const int idx = (lane < 16) ? lane : (lane - 24);
        float r = acc[0];
        r = (idx == 1) ? acc[1] : r;
        r = (idx == 2) ? acc[2] : r;
        r = (idx == 3) ? acc[3] : r;
        r = (idx == 4) ? acc[4] : r;
        r = (idx == 5) ? acc[5] : r;
        r = (idx == 6) ? acc[6] : r;
        r = (idx == 7) ? acc[7] : r;
        if (lane < 8 || lane >= 24) {
            const int pOut = (lane < 8) ? lane : (lane - 16); // point 0-7 / 8-15
            out[waveBase + 16 * g + pOut] = r;
        }
    }
}

extern "C" void kernel_launch(void* const* d_in, const int* in_sizes, int n_in,
                              void* d_out, int out_size, void* d_ws, size_t ws_size,
                              hipStream_t stream) {
    const float* coords = (const float*)d_in[0];   // [B,4] float32
    const float* mesh   = (const float*)d_in[1];   // [B,4096] float32
    float* out = (float*)d_out;                    // [B] float32

    const int batch = in_sizes[0] / 4;             // 32768
    const int block = 256;
    const int grid  = (batch + block - 1) / block; // 128 full blocks
    HL_interp4d_wmma<<<grid, block, 0, stream>>>(coords, mesh, out);
}